// CommonSpaceCache_33277406609685
// MI455X (gfx1250) — compile-verified
//
#include <hip/hip_runtime.h>
#include <hip/hip_bf16.h>

typedef __attribute__((ext_vector_type(2))) float v2f;
typedef __attribute__((ext_vector_type(8))) float v8f;

#define N_EMB   2048
#define DIM     768
#define NCACHE  32768
#define KSEL    3
#define SIM_THRESH 0.8f
#define EPSV    1e-8f

#define TM 64
#define TN 128
#define KC 32
#define LDP (KC + 4)   // 36-float row stride: bank-conflict-free AND 16B-aligned rows

// Async global->LDS B128 copy (CDNA5): per-lane LDS offset VGPR + 64-bit global addr.
__device__ __forceinline__ void async_copy_b128(void* lds_ptr, const float* gptr) {
  unsigned int lds_off = (unsigned int)(unsigned long long)lds_ptr; // low 32b = LDS offset
  unsigned long long ga = (unsigned long long)gptr;
  asm volatile("global_load_async_to_lds_b128 %0, %1, off"
               :: "v"(lds_off), "v"(ga) : "memory");
}
__device__ __forceinline__ void wait_async0() {
  asm volatile("s_wait_asynccnt 0x0" ::: "memory");
}

// ---------------- row norms (clamped at EPS, per reference) ----------------
__global__ __launch_bounds__(256) void row_norm_kernel(const float* __restrict__ x,
                                                       float* __restrict__ out) {
  __shared__ float red[256];
  int row = blockIdx.x;
  const float* p = x + (size_t)row * DIM;
  float s = 0.f;
  for (int j = threadIdx.x; j < DIM; j += 256) { float v = p[j]; s += v * v; }
  red[threadIdx.x] = s; __syncthreads();
  for (int st = 128; st > 0; st >>= 1) {
    if (threadIdx.x < st) red[threadIdx.x] += red[threadIdx.x + st];
    __syncthreads();
  }
  if (threadIdx.x == 0) out[row] = fmaxf(sqrtf(red[0]), EPSV);
}

// ------------- WMMA f32 GEMM: sim = (A @ B^T) / (|a||b|) -------------------
// Block: 256 threads = 8 waves; block tile 64(M) x 128(N); wave tile 32x32
// (2x2 of 16x16 -> 4 accumulators, 4 WMMA per 4 LDS fragment loads).
__global__ __launch_bounds__(256) void wmma_sim_kernel(const float* __restrict__ A,
                                                       const float* __restrict__ B,
                                                       const float* __restrict__ anorm,
                                                       const float* __restrict__ bnorm,
                                                       float* __restrict__ sim) {
  __shared__ float As[TM][LDP];
  __shared__ float Bs[TN][LDP];
  const int tid  = threadIdx.x;
  const int lane = tid & 31;
  const int w    = tid >> 5;
  const int wm   = w & 1;       // 2 wave-rows of 32
  const int wn   = w >> 1;      // 4 wave-cols of 32
  const int m0   = blockIdx.y * TM;
  const int n0   = blockIdx.x * TN;

  v8f acc00 = {}, acc01 = {}, acc10 = {}, acc11 = {};
  const int lm   = lane & 15;
  const int koff = (lane >> 4) * 2;   // ISA A/B frag: lanes 16-31 hold K+2,K+3
  const int am0  = wm * 32 + lm;      // A fragment rows (two tiles)
  const int am1  = am0 + 16;
  const int bn0  = wn * 32 + lm;      // B fragment rows (two tiles)
  const int bn1  = bn0 + 16;

  for (int kc = 0; kc < DIM; kc += KC) {
    // --- async global->LDS staging: A 64x32 (512 xB128), B 128x32 (1024 xB128) ---
    #pragma unroll
    for (int t = 0; t < 2; t++) {
      int s = t * 256 + tid;               // 0..511
      int r = s >> 3, c4 = s & 7;
      async_copy_b128(&As[r][c4 * 4], A + (size_t)(m0 + r) * DIM + kc + c4 * 4);
    }
    #pragma unroll
    for (int t = 0; t < 4; t++) {
      int s = t * 256 + tid;               // 0..1023
      int r = s >> 3, c4 = s & 7;
      async_copy_b128(&Bs[r][c4 * 4], B + (size_t)(n0 + r) * DIM + kc + c4 * 4);
    }
    if (kc + KC < DIM) { // prefetch next K slab -> global_prefetch_b8
      __builtin_prefetch(A + (size_t)(m0 + (tid >> 2)) * DIM + kc + KC, 0, 3);
      __builtin_prefetch(B + (size_t)(n0 + (tid >> 1)) * DIM + kc + KC, 0, 3);
    }
    wait_async0();
    __syncthreads();
    #pragma unroll
    for (int kk = 0; kk < KC; kk += 4) {
      v2f a0; a0.x = As[am0][kk + koff]; a0.y = As[am0][kk + koff + 1];
      v2f a1; a1.x = As[am1][kk + koff]; a1.y = As[am1][kk + koff + 1];
      v2f b0; b0.x = Bs[bn0][kk + koff]; b0.y = Bs[bn0][kk + koff + 1];
      v2f b1; b1.x = Bs[bn1][kk + koff]; b1.y = Bs[bn1][kk + koff + 1];
      acc00 = __builtin_amdgcn_wmma_f32_16x16x4_f32(false, a0, false, b0, (short)0, acc00, false, false);
      acc01 = __builtin_amdgcn_wmma_f32_16x16x4_f32(false, a0, false, b1, (short)0, acc01, false, false);
      acc10 = __builtin_amdgcn_wmma_f32_16x16x4_f32(false, a1, false, b0, (short)0, acc10, false, false);
      acc11 = __builtin_amdgcn_wmma_f32_16x16x4_f32(false, a1, false, b1, (short)0, acc11, false, false);
    }
    __syncthreads();
  }

  // Epilogue: C layout -> VGPR r: lanes0-15 M=r, lanes16-31 M=r+8; N = lane&15.
  const int col0  = n0 + wn * 32 + lm;
  const int col1  = col0 + 16;
  const float bnv0 = bnorm[col0];
  const float bnv1 = bnorm[col1];
  const int rb0   = m0 + wm * 32 + ((lane >> 4) * 8);
  const int rb1   = rb0 + 16;
  #pragma unroll
  for (int r = 0; r < 8; r++) {
    int row0 = rb0 + r, row1 = rb1 + r;
    float an0 = anorm[row0], an1 = anorm[row1];
    sim[(size_t)row0 * NCACHE + col0] = acc00[r] / (an0 * bnv0);
    sim[(size_t)row0 * NCACHE + col1] = acc01[r] / (an0 * bnv1);
    sim[(size_t)row1 * NCACHE + col0] = acc10[r] / (an1 * bnv0);
    sim[(size_t)row1 * NCACHE + col1] = acc11[r] / (an1 * bnv1);
  }
}

// ------- per-row argmax -> mask, accuracy counters, gather pos slot --------
__global__ __launch_bounds__(256) void argmax_pos_kernel(const float* __restrict__ sim,
                                                         const float* __restrict__ unknown_cache,
                                                         const int* __restrict__ labels,
                                                         const int* __restrict__ unknown_labels,
                                                         float* __restrict__ cand,
                                                         float* __restrict__ maskout,
                                                         unsigned int* __restrict__ counters) {
  __shared__ float bv[256]; __shared__ int bi[256]; __shared__ int sel;
  int row = blockIdx.x;
  const float* p = sim + (size_t)row * NCACHE;
  float best = -3.0e38f; int bidx = 0x7fffffff;
  for (int j = threadIdx.x; j < NCACHE; j += 256) {
    float v = p[j];
    if (v > best || (v == best && j < bidx)) { best = v; bidx = j; }
  }
  bv[threadIdx.x] = best; bi[threadIdx.x] = bidx; __syncthreads();
  for (int st = 128; st > 0; st >>= 1) {
    if (threadIdx.x < st) {
      float v2 = bv[threadIdx.x + st]; int i2 = bi[threadIdx.x + st];
      if (v2 > bv[threadIdx.x] || (v2 == bv[threadIdx.x] && i2 < bi[threadIdx.x])) {
        bv[threadIdx.x] = v2; bi[threadIdx.x] = i2;
      }
    }
    __syncthreads();
  }
  if (threadIdx.x == 0) {
    int idx = bi[0];
    int m = bv[0] > SIM_THRESH ? 1 : 0;
    maskout[row] = (float)m;
    if (m) {
      atomicAdd(&counters[0], 1u);
      if (labels[row] == unknown_labels[idx]) atomicAdd(&counters[1], 1u);
    }
    sel = idx;
  }
  __syncthreads();
  const float* src = unknown_cache + (size_t)sel * DIM;
  float* dst = cand + (size_t)row * 7 * DIM; // slot 0
  for (int j = threadIdx.x; j < DIM; j += 256) dst[j] = src[j];
}

// -- per-row rank-r select (r-th smallest, stable) via LDS radix histogram --
__global__ __launch_bounds__(256) void rank_select_kernel(const float* __restrict__ sim,
                                                          const float* __restrict__ unknown_cache,
                                                          const int* __restrict__ rand_idx,
                                                          float* __restrict__ cand) {
  extern __shared__ unsigned int smem[];
  unsigned int* keys = smem;             // 32768 keys (128 KB; CDNA5 WGP LDS = 320 KB)
  unsigned int* hist = smem + NCACHE;    // 256
  unsigned int* cnt  = hist + 256;       // 256
  __shared__ unsigned int s_sel, s_rr, s_idx;
  const int row = blockIdx.x, tid = threadIdx.x;
  const float* p = sim + (size_t)row * NCACHE;
  for (int j = tid; j < NCACHE; j += 256) {   // order-preserving f32 -> u32 map
    unsigned int u = __float_as_uint(p[j]);
    keys[j] = (u & 0x80000000u) ? ~u : (u | 0x80000000u);
  }
  __syncthreads();
  const int CH = NCACHE / 256;   // contiguous 128-element chunk per thread (keeps index order)
  const int c0 = tid * CH;
  for (int k = 0; k < KSEL; k++) {
    unsigned int prefix = 0, pmask = 0;
    if (tid == 0) s_rr = (unsigned int)rand_idx[row * KSEL + k];
    __syncthreads();
    for (int shift = 24; shift >= 0; shift -= 8) {
      hist[tid] = 0;
      __syncthreads();
      for (int j = c0; j < c0 + CH; j++) {
        unsigned int key = keys[j];
        if ((key & pmask) == prefix) atomicAdd(&hist[(key >> shift) & 255u], 1u);
      }
      __syncthreads();
      if (tid == 0) {
        unsigned int rr = s_rr, cum = 0; unsigned int selb = 0;
        for (int b = 0; b < 256; b++) {
          unsigned int h = hist[b];
          if (cum + h > rr) { selb = (unsigned int)b; break; }
          cum += h;
        }
        s_sel = selb; s_rr = rr - cum;
      }
      __syncthreads();
      prefix |= (s_sel << shift);
      pmask  |= (0xFFu << shift);
      __syncthreads();
    }
    // prefix = exact key; s_rr = rank among equal keys (stable -> index order)
    unsigned int c = 0;
    for (int j = c0; j < c0 + CH; j++) c += (keys[j] == prefix) ? 1u : 0u;
    cnt[tid] = c;
    __syncthreads();
    if (tid == 0) {
      unsigned int rr = s_rr, cum = 0;
      for (int t = 0; t < 256; t++) {
        if (cum + cnt[t] > rr) { s_sel = (unsigned int)t; s_rr = rr - cum; break; }
        cum += cnt[t];
      }
    }
    __syncthreads();
    if (tid == (int)s_sel) {
      unsigned int need = s_rr, seen = 0; int found = c0;
      for (int j = c0; j < c0 + CH; j++)
        if (keys[j] == prefix) { if (seen == need) { found = j; break; } seen++; }
      s_idx = (unsigned int)found;
    }
    __syncthreads();
    const float* src = unknown_cache + (size_t)s_idx * DIM;
    float* dst = cand + ((size_t)row * 7 + 1 + k) * DIM; // slots 1..3
    for (int j = tid; j < DIM; j += 256) dst[j] = src[j];
    __syncthreads();
  }
}

// ----------- per-row top-3 (desc value, asc index ties) + gather -----------
__global__ __launch_bounds__(256) void topk_known_kernel(const float* __restrict__ sim,
                                                         const float* __restrict__ known_cache,
                                                         float* __restrict__ cand) {
  __shared__ float sv[256 * 3]; __shared__ int si[256 * 3]; __shared__ int selidx[3];
  const int row = blockIdx.x, tid = threadIdx.x;
  const float* p = sim + (size_t)row * NCACHE;
  float tv[3] = {-3.0e38f, -3.0e38f, -3.0e38f};
  int   ti[3] = {0x7fffffff, 0x7fffffff, 0x7fffffff};
  for (int j = tid; j < NCACHE; j += 256) {
    float v = p[j];
    if (v > tv[0] || (v == tv[0] && j < ti[0])) {
      tv[2]=tv[1]; ti[2]=ti[1]; tv[1]=tv[0]; ti[1]=ti[0]; tv[0]=v; ti[0]=j;
    } else if (v > tv[1] || (v == tv[1] && j < ti[1])) {
      tv[2]=tv[1]; ti[2]=ti[1]; tv[1]=v; ti[1]=j;
    } else if (v > tv[2] || (v == tv[2] && j < ti[2])) {
      tv[2]=v; ti[2]=j;
    }
  }
  for (int t = 0; t < 3; t++) { sv[tid*3+t] = tv[t]; si[tid*3+t] = ti[t]; }
  __syncthreads();
  if (tid == 0) {
    float bv[3] = {-3.0e38f, -3.0e38f, -3.0e38f};
    int   bi2[3] = {0x7fffffff, 0x7fffffff, 0x7fffffff};
    for (int e = 0; e < 256 * 3; e++) {
      float v = sv[e]; int j = si[e];
      if (v > bv[0] || (v == bv[0] && j < bi2[0])) {
        bv[2]=bv[1]; bi2[2]=bi2[1]; bv[1]=bv[0]; bi2[1]=bi2[0]; bv[0]=v; bi2[0]=j;
      } else if (v > bv[1] || (v == bv[1] && j < bi2[1])) {
        bv[2]=bv[1]; bi2[2]=bi2[1]; bv[1]=v; bi2[1]=j;
      } else if (v > bv[2] || (v == bv[2] && j < bi2[2])) {
        bv[2]=v; bi2[2]=j;
      }
    }
    selidx[0]=bi2[0]; selidx[1]=bi2[1]; selidx[2]=bi2[2];
  }
  __syncthreads();
  for (int t = 0; t < 3; t++) {
    const float* src = known_cache + (size_t)selidx[t] * DIM;
    float* dst = cand + ((size_t)row * 7 + 4 + t) * DIM;  // slots 4..6
    for (int j = tid; j < DIM; j += 256) dst[j] = src[j];
  }
}

__global__ void init_counters_kernel(unsigned int* counters) {
  if (threadIdx.x < 2) counters[threadIdx.x] = 0u;
}

__global__ void finalize_kernel(const unsigned int* __restrict__ counters,
                                float* __restrict__ out_acc) {
  unsigned int nv = counters[0], ne = counters[1];
  out_acc[0] = (nv > 0) ? ((float)ne / (float)nv) : 0.0f;
}

extern "C" void kernel_launch(void* const* d_in, const int* in_sizes, int n_in,
                              void* d_out, int out_size, void* d_ws, size_t ws_size,
                              hipStream_t stream) {
  const float* embeddings     = (const float*)d_in[0];
  const float* known_cache    = (const float*)d_in[1];
  const float* unknown_cache  = (const float*)d_in[2];
  const int*   labels         = (const int*)d_in[3];
  const int*   unknown_labels = (const int*)d_in[5];
  const int*   rand_idx       = (const int*)d_in[6];

  float* out     = (float*)d_out;
  float* cand    = out;                             // [2048][7][768]
  float* maskout = out + (size_t)N_EMB * 7 * DIM;   // [2048]
  float* accout  = maskout + N_EMB;                 // [1]

  float* ws          = (float*)d_ws;
  float* emb_norm    = ws;
  float* unk_norm    = emb_norm + N_EMB;
  float* known_norm  = unk_norm + NCACHE;
  unsigned int* counters = (unsigned int*)(known_norm + NCACHE);
  float* simbuf      = (float*)(counters + 64);     // [2048][32768], reused for both caches

  init_counters_kernel<<<1, 32, 0, stream>>>(counters);
  row_norm_kernel<<<N_EMB,  256, 0, stream>>>(embeddings,    emb_norm);
  row_norm_kernel<<<NCACHE, 256, 0, stream>>>(unknown_cache, unk_norm);
  row_norm_kernel<<<NCACHE, 256, 0, stream>>>(known_cache,   known_norm);

  dim3 gemm_grid(NCACHE / TN, N_EMB / TM);

  // ---- unknown cache phase ----
  wmma_sim_kernel<<<gemm_grid, 256, 0, stream>>>(embeddings, unknown_cache,
                                                 emb_norm, unk_norm, simbuf);
  argmax_pos_kernel<<<N_EMB, 256, 0, stream>>>(simbuf, unknown_cache, labels,
                                               unknown_labels, cand, maskout, counters);
  size_t sel_lds = (size_t)(NCACHE + 256 + 256) * sizeof(unsigned int);
  rank_select_kernel<<<N_EMB, 256, sel_lds, stream>>>(simbuf, unknown_cache,
                                                      rand_idx, cand);

  // ---- known cache phase (simbuf reused; stream order serializes) ----
  wmma_sim_kernel<<<gemm_grid, 256, 0, stream>>>(embeddings, known_cache,
                                                 emb_norm, known_norm, simbuf);
  topk_known_kernel<<<N_EMB, 256, 0, stream>>>(simbuf, known_cache, cand);

  finalize_kernel<<<1, 1, 0, stream>>>(counters, accout);
}